// WindowSelfAttention_34772055228856
// MI455X (gfx1250) — compile-verified
//
#include <hip/hip_runtime.h>

// ---------------------------------------------------------------------------
// Window self-attention (Swin-style) for MI455X / gfx1250, fp32 end-to-end.
// Memory-bound (~1.8 GB traffic vs ~170 GFLOP) -> fp32 fidelity with
// V_WMMA_F32_16X16X4_F32 for all matmuls; async global->LDS staging for QKV.
// ---------------------------------------------------------------------------

typedef float v2f __attribute__((ext_vector_type(2)));
typedef float v8f __attribute__((ext_vector_type(8)));

// D = A(16x4 f32) * B(4x16 f32) + C(16x16 f32)
__device__ __forceinline__ v8f wmma_f32(v2f a, v2f b, v8f c) {
    return __builtin_amdgcn_wmma_f32_16x16x4_f32(false, a, false, b, (short)0, c,
                                                 false, false);
}

// Async copy one dword global -> LDS (GVS mode), tracked by ASYNCcnt.
__device__ __forceinline__ void async_load_f32(const float* base,
                                               unsigned global_byte_off,
                                               unsigned lds_byte_addr) {
    asm volatile("global_load_async_to_lds_b32 %0, %1, %2"
                 :
                 : "v"(lds_byte_addr), "v"(global_byte_off), "s"(base)
                 : "memory");
}

__device__ __forceinline__ void wait_async_all() {
    asm volatile("s_wait_asynccnt 0" ::: "memory");
}

#define HW      65536   // 256*256
#define NHEADS  8
#define HD      48
#define WTOK    64      // 8x8 window tokens
#define QKV_STR 49      // 48 + 1 pad  (coprime with 64 banks)
#define P_STR   65      // 64 + 1 pad

// ---------------------------------------------------------------------------
// GEMM: Y[b][m][hw] = sum_k W[m][k] * X[b][k][hw]      (channel-major in/out)
// Block = 8 waves; wave (mi,ni) computes 32 rows x 32 tokens (2x2 WMMA tiles).
// ---------------------------------------------------------------------------
__launch_bounds__(256)
__global__ void wmma_gemm_cmajor(const float* __restrict__ X,
                                 const float* __restrict__ W,
                                 float* __restrict__ Y,
                                 int K, int Mrows) {
    const int lane  = threadIdx.x & 31;
    const int wv    = threadIdx.x >> 5;      // 0..7
    const int mi    = wv & 3;                // 4 chunks of 32 rows -> 128/block
    const int ni    = wv >> 2;               // 2 chunks of 32 tokens -> 64/block
    const int t0    = blockIdx.x * 64 + ni * 32;   // global token base
    const int m0    = blockIdx.y * 128 + mi * 32;  // output-channel base
    const int b     = t0 >> 16;              // 65536 tokens per batch image
    const int hw0   = t0 & 65535;

    const float* Xb = X + (size_t)b * K * HW;
    float*       Yb = Y + (size_t)b * Mrows * HW;

    const int mlane = lane & 15;             // M (A) / N (B,D) position
    const int kb    = (lane >> 4) * 2;       // K pair held by this half-wave

    v8f acc[2][2] = {};
    for (int k0 = 0; k0 < K; k0 += 4) {
        v2f a[2], bb[2];
#pragma unroll
        for (int ms = 0; ms < 2; ++ms) {     // A: W row-major, float2 load
            const float* wp = W + (size_t)(m0 + ms * 16 + mlane) * K + k0 + kb;
            a[ms].x = wp[0];
            a[ms].y = wp[1];
        }
#pragma unroll
        for (int ns = 0; ns < 2; ++ns) {     // B: X channel-major, coalesced
            const float* xp = Xb + (size_t)(k0 + kb) * HW + hw0 + ns * 16 + mlane;
            bb[ns].x = xp[0];
            bb[ns].y = xp[HW];
        }
#pragma unroll
        for (int ms = 0; ms < 2; ++ms)
#pragma unroll
            for (int ns = 0; ns < 2; ++ns)
                acc[ms][ns] = wmma_f32(a[ms], bb[ns], acc[ms][ns]);
    }

    const int rowh = (lane >> 4) * 8;        // D layout: vgpr r -> row r / r+8
#pragma unroll
    for (int ms = 0; ms < 2; ++ms)
#pragma unroll
        for (int ns = 0; ns < 2; ++ns)
#pragma unroll
            for (int r = 0; r < 8; ++r)
                Yb[(size_t)(m0 + ms * 16 + rowh + r) * HW + hw0 + ns * 16 + mlane] =
                    acc[ms][ns][r];
}

// ---------------------------------------------------------------------------
// Window attention: one block (4 waves) per (window, head).
// S = QK^T * scale -> softmax (register shfl reductions) -> O = P*V.
// Q/K/V staged with GLOBAL_LOAD_ASYNC_TO_LDS_B32 (per-lane gather into
// padded LDS layout, no VGPR round-trip).
// ---------------------------------------------------------------------------
__launch_bounds__(128)
__global__ void window_attn(const float* __restrict__ QKV,
                            float* __restrict__ O) {
    __shared__ float qs[WTOK * QKV_STR];
    __shared__ float ks[WTOK * QKV_STR];
    __shared__ float vs[WTOK * QKV_STR];
    __shared__ float ps[WTOK * P_STR];

    const int head = blockIdx.x & 7;
    const int win  = blockIdx.x >> 3;
    const int b    = win >> 10;              // 32*32 = 1024 windows / image
    const int wrem = win & 1023;
    const int wh   = wrem >> 5;
    const int ww   = wrem & 31;
    const int hwb  = (wh * 8) * 256 + ww * 8;

    const size_t cbase = ((size_t)b * 1152 + head * HD) * HW;
    const float* Qg = QKV + cbase;
    const float* Kg = QKV + cbase + (size_t)384 * HW;
    const float* Vg = QKV + cbase + (size_t)768 * HW;

    // Async-stage Q/K/V tiles (64 tokens x 48 dims) into padded LDS.
    for (int i = threadIdx.x; i < WTOK * HD; i += 128) {
        const int t = i & 63;                // token in window
        const int d = i >> 6;                // head dim
        const unsigned gb = (unsigned)((d * HW + hwb + (t >> 3) * 256 + (t & 7)) * 4);
        const unsigned li = (unsigned)((t * QKV_STR + d) * 4);
        async_load_f32(Qg, gb, (unsigned)(uintptr_t)qs + li);
        async_load_f32(Kg, gb, (unsigned)(uintptr_t)ks + li);
        async_load_f32(Vg, gb, (unsigned)(uintptr_t)vs + li);
    }
    wait_async_all();
    __syncthreads();

    const int lane  = threadIdx.x & 31;
    const int wv    = threadIdx.x >> 5;      // wave handles q rows [16wv,16wv+16)
    const int mlane = lane & 15;
    const int kb    = (lane >> 4) * 2;
    const int qrow  = wv * 16;
    const int rowh  = (lane >> 4) * 8;

    // ---- S = Q * K^T  (M=16 q-tokens, N=64 k-tokens, K=48) ----
    v8f s[4] = {};
    for (int d0 = 0; d0 < HD; d0 += 4) {
        v2f a;
        a.x = qs[(qrow + mlane) * QKV_STR + d0 + kb];
        a.y = qs[(qrow + mlane) * QKV_STR + d0 + kb + 1];
#pragma unroll
        for (int n = 0; n < 4; ++n) {
            v2f bb;
            bb.x = ks[(n * 16 + mlane) * QKV_STR + d0 + kb];
            bb.y = ks[(n * 16 + mlane) * QKV_STR + d0 + kb + 1];
            s[n] = wmma_f32(a, bb, s[n]);
        }
    }

    // ---- softmax over 64 k-tokens, per q row (row = 16 lanes x 4 acc) ----
    const float scale = 0.14433756729740643f;   // 48^-0.5
#pragma unroll
    for (int r = 0; r < 8; ++r) {
        float m = s[0][r];
#pragma unroll
        for (int n = 1; n < 4; ++n) m = fmaxf(m, s[n][r]);
#pragma unroll
        for (int off = 1; off < 16; off <<= 1)
            m = fmaxf(m, __shfl_xor(m, off, 32));
        float sum = 0.f;
#pragma unroll
        for (int n = 0; n < 4; ++n) {
            const float p = __expf((s[n][r] - m) * scale);
            s[n][r] = p;
            sum += p;
        }
#pragma unroll
        for (int off = 1; off < 16; off <<= 1)
            sum += __shfl_xor(sum, off, 32);
        const float inv = __frcp_rn(sum);
#pragma unroll
        for (int n = 0; n < 4; ++n) s[n][r] *= inv;
    }

    // Re-layout P through LDS (D-layout -> A-layout).
#pragma unroll
    for (int n = 0; n < 4; ++n)
#pragma unroll
        for (int r = 0; r < 8; ++r)
            ps[(qrow + rowh + r) * P_STR + n * 16 + mlane] = s[n][r];
    __syncthreads();

    // ---- O = P * V  (M=16 q-tokens, N=48 dims, K=64 k-tokens) ----
    v8f o[3] = {};
    for (int k0 = 0; k0 < WTOK; k0 += 4) {
        v2f a;
        a.x = ps[(qrow + mlane) * P_STR + k0 + kb];
        a.y = ps[(qrow + mlane) * P_STR + k0 + kb + 1];
#pragma unroll
        for (int n = 0; n < 3; ++n) {
            v2f bb;
            bb.x = vs[(k0 + kb) * QKV_STR + n * 16 + mlane];
            bb.y = vs[(k0 + kb + 1) * QKV_STR + n * 16 + mlane];
            o[n] = wmma_f32(a, bb, o[n]);
        }
    }

    // Store to channel-major attention-output buffer [b][c=head*48+d][hw].
    float* Ob = O + ((size_t)b * 384 + head * HD) * HW;
#pragma unroll
    for (int n = 0; n < 3; ++n)
#pragma unroll
        for (int r = 0; r < 8; ++r) {
            const int d  = n * 16 + mlane;          // 0..47
            const int qt = qrow + rowh + r;         // token in window
            Ob[(size_t)d * HW + hwb + (qt >> 3) * 256 + (qt & 7)] = o[n][r];
        }
}

// ---------------------------------------------------------------------------
extern "C" void kernel_launch(void* const* d_in, const int* in_sizes, int n_in,
                              void* d_out, int out_size, void* d_ws, size_t ws_size,
                              hipStream_t stream) {
    const float* x      = (const float*)d_in[0];   // (2, 384, 256, 256)
    const float* w_qkv  = (const float*)d_in[1];   // (1152, 384)
    const float* w_proj = (const float*)d_in[2];   // (384, 384)
    float*       out    = (float*)d_out;           // (2, 384, 256, 256)

    float* qkv  = (float*)d_ws;                         // 2*1152*65536 floats
    float* attn = qkv + (size_t)2 * 1152 * HW;          // 2* 384*65536 floats

    const int Ttiles = (2 * HW) / 64;                   // 2048 token tiles

    // 1) qkv = w_qkv @ x        M=1152, K=384
    wmma_gemm_cmajor<<<dim3(Ttiles, 1152 / 128), dim3(256), 0, stream>>>(
        x, w_qkv, qkv, 384, 1152);

    // 2) windowed attention     2048 windows * 8 heads
    window_attn<<<dim3(2048 * NHEADS), dim3(128), 0, stream>>>(qkv, attn);

    // 3) out = w_proj @ attn    M=384, K=384
    wmma_gemm_cmajor<<<dim3(Ttiles, 384 / 128), dim3(256), 0, stream>>>(
        attn, w_proj, out, 384, 384);
}